// Varifold_loss_11596411699394
// MI455X (gfx1250) — compile-verified
//
#include <hip/hip_runtime.h>
#include <hip/hip_bf16.h>

typedef __attribute__((ext_vector_type(2))) float v2f;
typedef __attribute__((ext_vector_type(8))) float v8f;

#define SIGMA   0.03f
#define GAMMA   (1.0f / (SIGMA * SIGMA))
#define LOG2E   1.4426950408889634f
#define EPSF    1e-12f

static __device__ __forceinline__ float fast_exp2(float x) {
#if __has_builtin(__builtin_amdgcn_exp2f)
    return __builtin_amdgcn_exp2f(x);
#else
    return exp2f(x);
#endif
}

// ---------------------------------------------------------------------------
// Phase 1: per-face quantities for 2*B meshes, written in WMMA fragment order.
//
// Fragment layout per 16-face tile = 64 floats, stored as per-lane (v0,v1)
// PAIRS so phase 2 loads each fragment with a single global_load_b64:
//   lane l in 0..15 : buf[2*l+0] = x_l , buf[2*l+1] = y_l      (K=0, K=1)
//   lane l in 16..31: buf[2*l+0] = z_(l-16), buf[2*l+1] = 0    (K=2, K=3)
// This matches the 32-bit A(16x4) layout (v0: K=0 lanes0-15 / K=2 lanes16-31,
// v1: K=1 / K=3) and the mirrored B(4x16) layout.
// ---------------------------------------------------------------------------
__global__ void varifold_phase1(const float* __restrict__ preds,
                                const float* __restrict__ tgts,
                                const int*   __restrict__ faces,
                                float* __restrict__ CF,   // [2B][nTiles][64]
                                float* __restrict__ NFr,  // [2B][nTiles][64]
                                float* __restrict__ W,    // [2B][NFp]
                                int NV, int NF, int NFp, int B)
{
    int idx = blockIdx.x * blockDim.x + threadIdx.x;
    int total = 2 * B * NFp;
    if (idx >= total) return;

    int m = idx / NFp;
    int f = idx - m * NFp;
    int tile = f >> 4;
    int r    = f & 15;

    float cx = 0.f, cy = 0.f, cz = 0.f;
    float nx = 0.f, ny = 0.f, nz = 0.f;
    float w  = 0.f;

    if (f < NF) {
        const float* V = (m < B) ? (preds + (size_t)m * NV * 3)
                                 : (tgts  + (size_t)(m - B) * NV * 3);
        int i0 = faces[3 * f + 0];
        int i1 = faces[3 * f + 1];
        int i2 = faces[3 * f + 2];
        float ax = V[3 * i0 + 0], ay = V[3 * i0 + 1], az = V[3 * i0 + 2];
        float bx = V[3 * i1 + 0], by = V[3 * i1 + 1], bz = V[3 * i1 + 2];
        float gx = V[3 * i2 + 0], gy = V[3 * i2 + 1], gz = V[3 * i2 + 2];

        cx = (ax + bx + gx) * (1.0f / 3.0f);
        cy = (ay + by + gy) * (1.0f / 3.0f);
        cz = (az + bz + gz) * (1.0f / 3.0f);

        float e1x = bx - ax, e1y = by - ay, e1z = bz - az;
        float e2x = gx - ax, e2y = gy - ay, e2z = gz - az;
        float vx = 0.5f * (e1y * e2z - e1z * e2y);
        float vy = 0.5f * (e1z * e2x - e1x * e2z);
        float vz = 0.5f * (e1x * e2y - e1y * e2x);

        float L   = sqrtf(vx * vx + vy * vy + vz * vz);
        float inv = 1.0f / fmaxf(L, EPSF);
        nx = vx * inv; ny = vy * inv; nz = vz * inv;

        float c2 = cx * cx + cy * cy + cz * cz;
        w = L * fast_exp2(-(GAMMA * LOG2E) * c2);   // L * exp(-gamma*|C|^2)
    }

    size_t fb = ((size_t)m * (NFp >> 4) + tile) * 64;
    // lane r pair (x,y); lane 16+r pair (z,0)
    CF [fb + 2 * r + 0]        = cx;
    CF [fb + 2 * r + 1]        = cy;
    CF [fb + 2 * (16 + r) + 0] = cz;
    CF [fb + 2 * (16 + r) + 1] = 0.f;
    NFr[fb + 2 * r + 0]        = nx;
    NFr[fb + 2 * r + 1]        = ny;
    NFr[fb + 2 * (16 + r) + 0] = nz;
    NFr[fb + 2 * (16 + r) + 1] = 0.f;
    W[(size_t)m * NFp + f] = w;
}

// ---------------------------------------------------------------------------
// Phase 2: tiled 16x16 pair blocks via V_WMMA_F32_16X16X4_F32.
// grid.x = row tiles, grid.y = 3*B term instances; 8 waves/block stride columns.
// D[r][lane] of the 16x16 f32 accumulator maps to M = r + 8*(lane>=16), N = lane%16.
//
// `wave` is laundered through readfirstlane so the column-tile loop is a
// scalar (SGPR) loop: no per-iteration v_cmp / exec masking, and EXEC stays
// all-ones around the WMMAs. Prefetches use running per-lane pointers with a
// constant +0x800 folded into the instruction ioffset.
// ---------------------------------------------------------------------------
__global__ __launch_bounds__(256)
void varifold_phase2(const float* __restrict__ CF,
                     const float* __restrict__ NFr,
                     const float* __restrict__ W,
                     float* __restrict__ out,
                     int nTiles, int NFp, int B)
{
    int rowTile = blockIdx.x;
    int term    = blockIdx.y;            // 0..3*B-1
    int b       = term / 3;
    int kind    = term - 3 * b;          // 0: (s,s)+1   1: (t,t)+1   2: (s,t)-2
    int mx      = (kind == 1) ? (B + b) : b;
    int my      = (kind == 0) ? b : (B + b);
    float coef  = ((kind == 2) ? -2.0f : 1.0f) / (float)B;

    int lane = threadIdx.x & 31;
    // uniform per wave -> force into an SGPR so the loop below is scalar
    int wave = __builtin_amdgcn_readfirstlane(threadIdx.x >> 5);
    int l15  = lane & 15;
    int hi   = lane >> 4;                // 0 => rows M=0..7, 1 => rows M=8..15

    const float* CFx = CF  + (size_t)mx * nTiles * 64;
    const float* NFx = NFr + (size_t)mx * nTiles * 64;
    const float* Wx  = W   + (size_t)mx * NFp;
    const float* CFy = CF  + (size_t)my * nTiles * 64;
    const float* NFy = NFr + (size_t)my * nTiles * 64;
    const float* Wy  = W   + (size_t)my * NFp;

    // A fragments for this row tile; fold 2*gamma*log2(e) into A so that
    // P = 2*gamma*log2e*(Cx.Cy) and exp2(P) = exp(2*gamma*Cx.Cy).
    const float ascale = 2.0f * GAMMA * LOG2E;
    v2f aC = *(const v2f*)(CFx + (size_t)rowTile * 64 + 2 * lane);
    v2f aN = *(const v2f*)(NFx + (size_t)rowTile * 64 + 2 * lane);
    aC.x *= ascale;
    aC.y *= ascale;

    float wxv[8];
#pragma unroll
    for (int r = 0; r < 8; ++r)
        wxv[r] = Wx[rowTile * 16 + r + 8 * hi];   // wx for row M = r + 8*hi

    // running per-lane pointers for this wave's column tiles (stride 8 tiles)
    const float* pc = CFy + (size_t)wave * 64 + 2 * lane;
    const float* pn = NFy + (size_t)wave * 64 + 2 * lane;
    const float* pw = Wy  + wave * 16 + l15;

    int nIter = (nTiles > wave) ? ((nTiles - wave + 7) >> 3) : 0;  // scalar

    float acc = 0.f;
    for (int it = 0; it < nIter; ++it) {
        v2f   bC = *(const v2f*)pc;
        v2f   bN = *(const v2f*)pn;
        float wy = *pw;                           // wy for col N = lane%16

        // prefetch next column tile; +0x800 folds into the prefetch ioffset
        __builtin_prefetch(pc + 512, 0, 3);
        __builtin_prefetch(pn + 512, 0, 3);

        v8f zero = {};
        v8f P = __builtin_amdgcn_wmma_f32_16x16x4_f32(
                    false, aC, false, bC, (short)0, zero, false, false);
        v8f Q = __builtin_amdgcn_wmma_f32_16x16x4_f32(
                    false, aN, false, bN, (short)0, zero, false, false);

        float ts = 0.f;
#pragma unroll
        for (int r = 0; r < 8; ++r) {
            float e = fast_exp2(P[r]);            // exp(2*gamma*Ci.Cj)
            float q = Q[r];
            ts = __builtin_fmaf(wxv[r] * e, q * q, ts);
        }
        acc = __builtin_fmaf(wy, ts, acc);

        pc += 8 * 64;
        pn += 8 * 64;
        pw += 8 * 16;
    }

    // wave32 reduction
#pragma unroll
    for (int off = 16; off > 0; off >>= 1)
        acc += __shfl_down(acc, off, 32);

    if (lane == 0)
        atomicAdd(out, coef * acc);
}

__global__ void varifold_zero(float* __restrict__ out, int n)
{
    int i = blockIdx.x * blockDim.x + threadIdx.x;
    if (i < n) out[i] = 0.f;
}

extern "C" void kernel_launch(void* const* d_in, const int* in_sizes, int n_in,
                              void* d_out, int out_size, void* d_ws, size_t ws_size,
                              hipStream_t stream)
{
    const float* preds = (const float*)d_in[0];
    const float* tgts  = (const float*)d_in[1];
    const int*   faces = (const int*)d_in[2];
    float*       out   = (float*)d_out;

    const int B   = 4;
    const int NF  = in_sizes[2] / 3;
    const int NV  = in_sizes[0] / (3 * B);
    const int NFp = (NF + 15) & ~15;
    const int nTiles = NFp >> 4;

    // workspace partition (floats)
    size_t cfElems = (size_t)2 * B * nTiles * 64;
    float* CF  = (float*)d_ws;
    float* NFr = CF + cfElems;
    float* W   = NFr + cfElems;
    (void)ws_size;

    // zero the output accumulator
    varifold_zero<<<1, 64, 0, stream>>>(out, out_size);

    // phase 1: face quantities in fragment order
    {
        int total = 2 * B * NFp;
        int blk = 256;
        varifold_phase1<<<(total + blk - 1) / blk, blk, 0, stream>>>(
            preds, tgts, faces, CF, NFr, W, NV, NF, NFp, B);
    }

    // phase 2: pairwise tiles via WMMA
    {
        dim3 grid(nTiles, 3 * B);
        varifold_phase2<<<grid, 256, 0, stream>>>(CF, NFr, W, out, nTiles, NFp, B);
    }
}